// MPS_RNN_1D_3633542332988
// MI455X (gfx1250) — compile-verified
//
#include <hip/hip_runtime.h>
#include <math.h>

typedef float v2f __attribute__((ext_vector_type(2)));
typedef float v8f __attribute__((ext_vector_type(8)));

#define NQ   64   // qubits
#define LQ   32   // sites = NQ/2
#define DC   32   // DCUT
#define HL   4    // HLOC
#define BT   16   // batch elements per wave
#define HPAD 34   // padded LDS row (bank spread)

__global__ __launch_bounds__(32)
void mps_rnn_1d_kernel(const int*   __restrict__ X,     // (B, 64) in {-1,+1}
                       const float* __restrict__ Mg,    // (L, HL, DC, DC)
                       const float* __restrict__ Vg,    // (L, HL, DC)
                       const float* __restrict__ Wg,    // (L, DC)
                       const float* __restrict__ Cg,    // (L,)
                       float*       __restrict__ out)   // (B,) complex64 interleaved
{
    __shared__ float hp[BT][HPAD];   // carry vectors y_t/||y|| per batch element
    __shared__ int   t_lds[BT];
    __shared__ int   q_lds[BT];

    const int lane  = threadIdx.x;   // 0..31 (wave32)
    const int half  = lane >> 4;     // 0/1
    const int l15   = lane & 15;
    const int mbase = blockIdx.x * BT;

    // h0 = ones -> hp0 = ones (independent of q at step 0)
    for (int j = lane; j < BT * DC; j += 32)
        hp[j >> 5][j & 31] = 1.0f;

    float myAmp = 1.0f;   // valid on owner lanes (l15 < 8)
    float myPhi = 0.0f;
    int   prevT = 0;

    __syncthreads();

    for (int n = 0; n < LQ; ++n) {
        // ---- per-step selection codes t=idx[n], q=idx[n-1] (q=t at n=0)
        if (lane < BT) {
            const int row = (mbase + lane) * NQ + 2 * n;
            const int x0 = X[row];
            const int x1 = X[row + 1];
            const int tt = ((x0 + 1) >> 1) | (((x1 + 1) >> 1) << 1);
            t_lds[lane] = tt;
            q_lds[lane] = (n == 0) ? tt : prevT;
            prevT = tt;
        }
        __syncthreads();

        const float w0 = Wg[n * DC + l15];
        const float w1 = Wg[n * DC + 16 + l15];
        const float cn = Cg[n];
        const float* Mn = Mg + n * (HL * DC * DC);
        const float* Vn = Vg + n * (HL * DC);

        // ---- accumulators initialized with bias v[i][b] (same for every batch row)
        v8f acc[HL][2];
#pragma unroll
        for (int i = 0; i < HL; ++i) {
#pragma unroll
            for (int nb = 0; nb < 2; ++nb) {
                const float bias = Vn[i * DC + nb * 16 + l15];
                v8f c;
#pragma unroll
                for (int r = 0; r < 8; ++r) c[r] = bias;
                acc[i][nb] = c;
            }
        }

        // ---- Y[i] = HP(16x32) @ M[i](32x32) + v[i] via V_WMMA_F32_16X16X4_F32
        // A frag: lane m=l15 rows, VGPR0:K=k0+2*half, VGPR1:K=k0+2*half+1
        // B frag: lane n=l15 cols, same K split
#pragma unroll
        for (int kc = 0; kc < 8; ++kc) {
            const int k0 = kc * 4 + 2 * half;
            const v2f a2 = *(const v2f*)&hp[l15][k0];   // 8B aligned (k0 even, row 136B)
#pragma unroll
            for (int i = 0; i < HL; ++i) {
                const float* Mi = Mn + i * (DC * DC) + k0 * DC;
#pragma unroll
                for (int nb = 0; nb < 2; ++nb) {
                    v2f b2;
                    b2.x = Mi[nb * 16 + l15];        // row k0+2*half
                    b2.y = Mi[DC + nb * 16 + l15];   // row k0+2*half+1
                    acc[i][nb] = __builtin_amdgcn_wmma_f32_16x16x4_f32(
                        false, a2, false, b2, (short)0, acc[i][nb], false, false);
                }
            }
        }

        // ---- per-batch reductions:  nrm^2 = sum_i ||y_i||^2 ;
        //      amp *= sqrt(||y_t||^2 / nrm^2) ; phi += pi if (y_q.w + c*nrm < 0)
        //      carry hp = y_t / nrm
        // D layout: acc[i][nb][r] @ lane = Y[i, batch r+8*half, nb*16+l15]
#pragma unroll
        for (int r = 0; r < 8; ++r) {
            const int m  = r + 8 * half;
            const int it = t_lds[m];
            const int iq = q_lds[m];
            float sp = 0.f, st = 0.f, dq = 0.f;
            float v0 = 0.f, v1 = 0.f;
#pragma unroll
            for (int i = 0; i < HL; ++i) {
                const float e0 = acc[i][0][r];
                const float e1 = acc[i][1][r];
                const float ss = e0 * e0 + e1 * e1;
                sp += ss;
                if (i == it) { st = ss; v0 = e0; v1 = e1; }
                if (i == iq) { dq = e0 * w0 + e1 * w1; }
            }
            // reduce across the 16-lane half (masks <= 8 never cross halves)
#pragma unroll
            for (int mk = 1; mk <= 8; mk <<= 1) {
                sp += __shfl_xor(sp, mk, 32);
                st += __shfl_xor(st, mk, 32);
                dq += __shfl_xor(dq, mk, 32);
            }
            const float nrm = sqrtf(sp);
            const float inv = 1.0f / nrm;
            if (l15 == r) {   // owner lanes: lane r (batch r), lane r+16 (batch r+8)
                myAmp *= sqrtf(st) * inv;
                if (dq + cn * nrm < 0.0f) myPhi += 3.14159265358979323846f;
            }
            hp[m][l15]      = v0 * inv;
            hp[m][l15 + 16] = v1 * inv;
        }
        __syncthreads();
    }

    // owner lanes write amp * exp(i*phi), interleaved complex64
    if (l15 < 8) {
        const int mg = mbase + l15 + 8 * half;
        out[2 * mg]     = myAmp * cosf(myPhi);
        out[2 * mg + 1] = myAmp * sinf(myPhi);
    }
}

extern "C" void kernel_launch(void* const* d_in, const int* in_sizes, int n_in,
                              void* d_out, int out_size, void* d_ws, size_t ws_size,
                              hipStream_t stream) {
    const int*   X  = (const int*)  d_in[0];   // x
    const float* Mg = (const float*)d_in[1];   // parm_M
    const float* Vg = (const float*)d_in[2];   // parm_v
    const float* Wg = (const float*)d_in[3];   // parm_w
    const float* Cg = (const float*)d_in[4];   // parm_c
    // d_in[5] = parm_eta: constant -> gamma = eta*I, cancels exactly (see analysis)

    const int batch = in_sizes[0] / NQ;        // 4096
    dim3 grid(batch / BT), block(32);
    mps_rnn_1d_kernel<<<grid, block, 0, stream>>>(X, Mg, Vg, Wg, Cg, (float*)d_out);
}